// EEGTrainAugment_34351148433697
// MI455X (gfx1250) — compile-verified
//
#include <hip/hip_runtime.h>

// EEG train-time augmentation: channel dropout + gaussian noise + temporal mask.
// Pure streaming kernel: 12 B of HBM traffic per element, ~3 flops.
// MI455X roofline: 805 MB / 23.3 TB/s ~= 35 us. Memory-bound -> b128 NT stream.

typedef float v4f __attribute__((ext_vector_type(4)));

constexpr int   kB        = 256;
constexpr int   kT        = 2048;
constexpr int   kN        = 128;
constexpr float kPChDrop  = 0.1f;
constexpr float kNoiseStd = 0.02f;
constexpr float kPTimeMsk = 0.1f;
constexpr int   kMaxMask  = 32;            // min(MAX_MASK, T) == 32

constexpr int kThreads    = 256;           // 8 wave32s per block
constexpr int kIters      = 8;             // float4s per thread (unrolled)
constexpr int kF4PerB     = kT * kN / 4;   // 65536 float4s per batch sample
constexpr int kF4PerBlk   = kThreads * kIters;   // 2048
constexpr int kBlocksX    = kF4PerB / kF4PerBlk; // 32
constexpr int kTPerBlk    = kF4PerBlk * 4 / kN;  // 64 time rows per block
constexpr int kTPerIter   = kThreads * 4 / kN;   // 8 time rows per iteration

__global__ __launch_bounds__(kThreads)
void eeg_aug_kernel(const float* __restrict__ xg,
                    const float* __restrict__ chg,
                    const float* __restrict__ nzg,
                    const float* __restrict__ tmg,
                    const int*   __restrict__ t0g,
                    float*       __restrict__ outg)
{
    const int b   = blockIdx.y;     // batch sample
    const int tid = threadIdx.x;    // 0..255

    const v4f* __restrict__ x   = (const v4f*)xg;
    const v4f* __restrict__ ch  = (const v4f*)chg;
    const v4f* __restrict__ nz  = (const v4f*)nzg;
    v4f*       __restrict__ out = (v4f*)outg;

    // --- Channel-drop keep mask: the channel pattern repeats every 128 elems
    // (= 32 float4s = one wave), and the per-iteration stride (1024 elems) is a
    // multiple of N, so each thread's 4 channels are fixed. Load once.
    const v4f cu = ch[b * (kN / 4) + (tid & 31)];
    v4f keep;
    keep.x = (cu.x < kPChDrop) ? 0.0f : 1.0f;
    keep.y = (cu.y < kPChDrop) ? 0.0f : 1.0f;
    keep.z = (cu.z < kPChDrop) ? 0.0f : 1.0f;
    keep.w = (cu.w < kPChDrop) ? 0.0f : 1.0f;

    // --- Per-sample temporal-mask parameters (wave-uniform scalar loads).
    const bool apply = tmg[b] < kPTimeMsk;
    const int  ts    = t0g[b];
    const int  te    = ts + kMaxMask;

    // --- Flat float4 index; within a wave all 32 lanes share one time row.
    const long long base =
        (long long)b * kF4PerB + (long long)blockIdx.x * kF4PerBlk + tid;
    const int tbase = blockIdx.x * kTPerBlk + (tid >> 5);

    // Prefetch the tail of this block's stream into cache ahead of the loads.
    __builtin_prefetch(x  + base + 4 * kThreads, 0, 0);
    __builtin_prefetch(nz + base + 4 * kThreads, 0, 0);

#pragma unroll
    for (int i = 0; i < kIters; ++i) {
        const long long idx = base + (long long)i * kThreads;

        // Streaming loads: zero reuse, 768 MB total > 192 MB L2 -> non-temporal.
        v4f xv = __builtin_nontemporal_load(x  + idx);
        v4f nv = __builtin_nontemporal_load(nz + idx);

        const int   t     = tbase + i * kTPerIter;         // wave-uniform
        const float tkeep = (apply && t >= ts && t < te) ? 0.0f : 1.0f;

        v4f r = (xv * keep + kNoiseStd * nv) * tkeep;

        __builtin_nontemporal_store(r, out + idx);
    }
}

extern "C" void kernel_launch(void* const* d_in, const int* in_sizes, int n_in,
                              void* d_out, int out_size, void* d_ws, size_t ws_size,
                              hipStream_t stream) {
    (void)in_sizes; (void)n_in; (void)out_size; (void)d_ws; (void)ws_size;

    const float* x  = (const float*)d_in[0];  // [B,T,N]
    const float* ch = (const float*)d_in[1];  // [B,1,N]
    const float* nz = (const float*)d_in[2];  // [B,T,N]
    const float* tm = (const float*)d_in[3];  // [B]
    const int*   t0 = (const int*)  d_in[4];  // [B]
    float*       o  = (float*)d_out;          // [B,T,N]

    dim3 grid(kBlocksX, kB, 1);   // (32, 256)
    dim3 block(kThreads, 1, 1);   // 256 threads = 8 wave32s
    eeg_aug_kernel<<<grid, block, 0, stream>>>(x, ch, nz, tm, t0, o);
}